// CRFLayer_18571438587915
// MI455X (gfx1250) — compile-verified
//
#include <hip/hip_runtime.h>
#include <hip/hip_bf16.h>

// ---- problem constants (match reference) ----
#define Bb   128
#define Ll   1024
#define Hh   512
#define Tt   64
#define BEGIN_TAG 0
#define END_TAG   1

typedef __attribute__((ext_vector_type(2))) float v2f;
typedef __attribute__((ext_vector_type(8))) float v8f;

// =====================================================================
// Kernel 1: emissions = hiddens @ W + b   via V_WMMA_F32_16X16X4_F32
//   hiddens: (M=131072, 512) fp32 row-major
//   W:       (512, 64)       fp32 row-major
//   out:     (M, 64)         fp32
// 256 threads = 8 waves/block; each wave computes a 16x64 strip.
// =====================================================================
__global__ __launch_bounds__(256) void emis_gemm_wmma(
    const float* __restrict__ A,
    const float* __restrict__ W,
    const float* __restrict__ bias,
    float* __restrict__ out)
{
    const int wave = threadIdx.x >> 5;
    const int lane = threadIdx.x & 31;
    const int half = lane >> 4;   // 0: K pair {0,1}, 1: K pair {2,3}
    const int lidx = lane & 15;   // row (A) / col (B) within tile
    const long mbase = ((long)blockIdx.x * 8 + wave) * 16;

    v8f c0 = {}, c1 = {}, c2 = {}, c3 = {};

    const float* arow = A + (mbase + lidx) * Hh;

    for (int k0 = 0; k0 < Hh; k0 += 4) {
        // A fragment: lane holds A[row][k0+2*half .. +1]  (8B aligned)
        v2f a = *reinterpret_cast<const v2f*>(arow + k0 + 2 * half);

        // B fragments from W: b.x = W[kb][col], b.y = W[kb+1][col]
        const int kb = k0 + 2 * half;
        const float* w0 = W + kb * Tt + lidx;
        v2f b;
        b.x = w0[0];   b.y = w0[Tt];
        c0 = __builtin_amdgcn_wmma_f32_16x16x4_f32(false, a, false, b, (short)0, c0, false, false);
        b.x = w0[16];  b.y = w0[Tt + 16];
        c1 = __builtin_amdgcn_wmma_f32_16x16x4_f32(false, a, false, b, (short)0, c1, false, false);
        b.x = w0[32];  b.y = w0[Tt + 32];
        c2 = __builtin_amdgcn_wmma_f32_16x16x4_f32(false, a, false, b, (short)0, c2, false, false);
        b.x = w0[48];  b.y = w0[Tt + 48];
        c3 = __builtin_amdgcn_wmma_f32_16x16x4_f32(false, a, false, b, (short)0, c3, false, false);
    }

    // C layout: lanes 0-15 -> N=lane, M=v ; lanes 16-31 -> N=lane-16, M=v+8
    const int col = lidx;
    const float bv0 = bias[col];
    const float bv1 = bias[col + 16];
    const float bv2 = bias[col + 32];
    const float bv3 = bias[col + 48];
#pragma unroll
    for (int v = 0; v < 8; ++v) {
        const long row = mbase + v + 8 * half;
        float* o = out + row * Tt + col;
        o[0]  = c0[v] + bv0;
        o[16] = c1[v] + bv1;
        o[32] = c2[v] + bv2;
        o[48] = c3[v] + bv3;
    }
}

// =====================================================================
// Kernel 2: forward (partition) scan per batch. 1 block = 1 batch,
// 64 threads (thread i owns state i). E-row in registers; per step:
//   M = max_j fv[j];  p[j]=exp(fv[j]-M);  s_i = dot(E[i,:], p);
//   fv[i] = log(s_i) + M + em[t,i]
// =====================================================================
__global__ __launch_bounds__(64) void crf_scan(
    const float* __restrict__ emis,   // (B*L, 64)
    const int*   __restrict__ lens,
    const float* __restrict__ trans,  // (64, 64): trans[i][j] = FROM j TO i
    float* __restrict__ fwd)          // (B,)
{
    __shared__ __align__(16) float fv[Tt];
    __shared__ __align__(16) float p[Tt];

    const int b = blockIdx.x;
    const int i = threadIdx.x;

    // Precompute exp(transition[i, :]) into registers (constant over scan).
    float e[Tt];
#pragma unroll
    for (int j = 0; j < Tt; ++j)
        e[j] = __expf(trans[i * Tt + j]);

    fv[i] = trans[i * Tt + BEGIN_TAG];   // init = transition[:, BEGIN]
    const int len = lens[b];
    const float* em = emis + (size_t)b * Ll * Tt;
    __syncthreads();

    for (int t = 0; t < len; ++t) {
        // M = max_j fv[j] (broadcast LDS reads, vectorized)
        float M = -3.0e38f;
        const float4* fv4 = reinterpret_cast<const float4*>(fv);
#pragma unroll
        for (int j = 0; j < Tt / 4; ++j) {
            float4 v = fv4[j];
            M = fmaxf(M, fmaxf(fmaxf(v.x, v.y), fmaxf(v.z, v.w)));
        }
        p[i] = __expf(fv[i] - M);
        const float emv = em[(size_t)t * Tt + i];
        if (t + 1 < len)
            __builtin_prefetch(&em[(size_t)(t + 1) * Tt + i], 0, 3);
        __syncthreads();

        float s = 0.0f;
#pragma unroll
        for (int j = 0; j < Tt; ++j)
            s = __builtin_fmaf(p[j], e[j], s);

        fv[i] = __logf(s) + M + emv;
        __syncthreads();
    }

    // fwd = logsumexp_i(fv[i] + transition[END, i])
    p[i] = fv[i] + trans[END_TAG * Tt + i];
    __syncthreads();
    if (i == 0) {
        float M = -3.0e38f;
#pragma unroll
        for (int j = 0; j < Tt; ++j) M = fmaxf(M, p[j]);
        float s = 0.0f;
#pragma unroll
        for (int j = 0; j < Tt; ++j) s += __expf(p[j] - M);
        fwd[b] = __logf(s) + M;
    }
}

// =====================================================================
// Kernel 3: gold path score + final combine: out[b] = fwd[b] - gold[b]
// =====================================================================
__global__ __launch_bounds__(256) void gold_combine(
    const float* __restrict__ emis,   // (B*L, 64)
    const int*   __restrict__ lens,
    const int*   __restrict__ tags,   // (B, L)
    const float* __restrict__ trans,  // (64, 64)
    const float* __restrict__ fwd,    // (B,)
    float* __restrict__ out)          // (B,)
{
    __shared__ float red[256];
    const int b = blockIdx.x;
    const int tid = threadIdx.x;
    const int len = lens[b];
    const int* tg = tags + (size_t)b * Ll;
    const float* em = emis + (size_t)b * Ll * Tt;

    float acc = 0.0f;
    for (int t = tid; t < len; t += 256) {
        const int tag  = tg[t];
        const int prev = (t == 0) ? BEGIN_TAG : tg[t - 1];
        acc += trans[tag * Tt + prev] + em[(size_t)t * Tt + tag];
    }
    red[tid] = acc;
    __syncthreads();
#pragma unroll
    for (int s = 128; s > 0; s >>= 1) {
        if (tid < s) red[tid] += red[tid + s];
        __syncthreads();
    }
    if (tid == 0) {
        const int last = tg[len - 1];
        const float gold = red[0] + trans[END_TAG * Tt + last];
        out[b] = fwd[b] - gold;
    }
}

extern "C" void kernel_launch(void* const* d_in, const int* in_sizes, int n_in,
                              void* d_out, int out_size, void* d_ws, size_t ws_size,
                              hipStream_t stream) {
    const float* hiddens = (const float*)d_in[0];   // (B, L, H) f32
    const int*   lens    = (const int*)  d_in[1];   // (B,)      i32
    const int*   tags    = (const int*)  d_in[2];   // (B, L)    i32
    const float* W       = (const float*)d_in[3];   // (H, T)    f32
    const float* bias    = (const float*)d_in[4];   // (T,)      f32
    const float* trans   = (const float*)d_in[5];   // (T, T)    f32
    float* out = (float*)d_out;                     // (B,)      f32

    float* emis = (float*)d_ws;                          // 131072*64 f32 = 32 MB
    float* fwd  = emis + (size_t)Bb * Ll * Tt;           // 128 f32

    // 1) emissions GEMM: M=131072 rows, 8 waves * 16 rows per block
    emis_gemm_wmma<<<dim3((Bb * Ll) / (16 * 8)), 256, 0, stream>>>(hiddens, W, bias, emis);
    // 2) sequential forward scan, one block per batch
    crf_scan<<<dim3(Bb), Tt, 0, stream>>>(emis, lens, trans, fwd);
    // 3) gold score + combine
    gold_combine<<<dim3(Bb), 256, 0, stream>>>(emis, lens, tags, trans, fwd, out);
}